// DistillationLoss_10290741641679
// MI455X (gfx1250) — compile-verified
//
#include <hip/hip_runtime.h>
#include <math.h>

// ---------------------------------------------------------------------------
// DistillationLoss on MI455X (gfx1250, wave32)
//
// Roofline: dominant cost = streaming 8192x8192 f32 logits (256 MiB) once
// at 23.3 TB/s ~= 11.5us. exp count ~67M is well under that even at 1/4-rate
// transcendental throughput -> HBM-bound. Single-pass online softmax per row,
// float4 (global_load_b128) loads, 1 workgroup (8 wave32) per row.
// Sparse KL terms fused in the same block (scattered reloads hit L1/L2).
// Final 8192 -> 1 reduction done with V_WMMA_F32_16X16X4_F32 (ones-vector
// matmul trick, f32 end to end), check-free inner loop.
// ---------------------------------------------------------------------------

#define NGLOB 16384
#define TEMP 2.0f
#define KMAX 64

typedef __attribute__((ext_vector_type(2))) float v2f;
typedef __attribute__((ext_vector_type(8))) float v8f;

// ---- g2l table: init to -1 -------------------------------------------------
__global__ void g2l_init_kernel(int* __restrict__ g2l) {
    int i = blockIdx.x * blockDim.x + threadIdx.x;
    if (i < NGLOB) g2l[i] = -1;
}

// ---- g2l table: scatter batch_indices -> local row ids ---------------------
__global__ void g2l_scatter_kernel(const int* __restrict__ bidx,
                                   int* __restrict__ g2l, int Bn) {
    int j = blockIdx.x * blockDim.x + threadIdx.x;
    if (j < Bn) {
        int g = bidx[j];
        if (g >= 0 && g < NGLOB) g2l[g] = j;
    }
}

// ---- per-row: online log-sum-exp + sparse KL terms -------------------------
__global__ void __launch_bounds__(256)
row_loss_kernel(const float* __restrict__ logits,
                const int* __restrict__ tIdx,
                const float* __restrict__ tScore,
                const int* __restrict__ g2l,
                float* __restrict__ rowLoss,
                int Bn, int K) {
    const int row = blockIdx.x;
    const int tid = threadIdx.x;
    const float invT = 1.0f / TEMP;
    const float* rowp = logits + (size_t)row * (size_t)Bn;

    // ---- streaming online softmax (x = logits*invT) ----
    float m = -3.402823466e38f;  // running max
    float s = 0.0f;              // running sum of exp(x - m)
    for (int base = tid * 4; base < Bn; base += 256 * 4) {
        // speculative prefetch of next sweep; safe past end (ISA 10.5)
        __builtin_prefetch((const void*)(rowp + base + 1024), 0, 1);
        float4 v = *(const float4*)(rowp + base);
        float x0 = v.x * invT, x1 = v.y * invT, x2 = v.z * invT, x3 = v.w * invT;
        float mx = fmaxf(fmaxf(x0, x1), fmaxf(x2, x3));
        if (mx > m) { s *= __expf(m - mx); m = mx; }
        s += __expf(x0 - m) + __expf(x1 - m) + __expf(x2 - m) + __expf(x3 - m);
    }

    __shared__ float sm[256];
    __shared__ float ss[256];
    sm[tid] = m; ss[tid] = s;
    __syncthreads();
    for (int off = 128; off > 0; off >>= 1) {
        if (tid < off) {
            float m1 = sm[tid], s1 = ss[tid];
            float m2 = sm[tid + off], s2 = ss[tid + off];
            float M = fmaxf(m1, m2);
            ss[tid] = s1 * __expf(m1 - M) + s2 * __expf(m2 - M);
            sm[tid] = M;
        }
        __syncthreads();
    }
    const float c = sm[0] + __logf(ss[0]);  // logsumexp; logp_j = x_j - c
    __syncthreads();

    // ---- sparse target: remap, dedup (last-write-wins), diag=1, normalize --
    __shared__ int   sl[KMAX];
    __shared__ float sc[KMAX];
    __shared__ float red[KMAX];
    if (tid < KMAX) {
        int l = -1; float sv = 0.0f;
        if (tid < K) {
            int ti = tIdx[(size_t)row * K + tid];
            if (ti >= 0 && ti < NGLOB) l = g2l[ti];
            sv = tScore[(size_t)row * K + tid];
        }
        sl[tid] = l; sc[tid] = sv;
    }
    __syncthreads();

    bool active = false;
    if (tid < KMAX) {
        int l = sl[tid];
        if (l >= 0 && l != row && tid < K) {
            active = true;
            for (int k2 = tid + 1; k2 < K; ++k2)       // .set scatter: last wins
                if (sl[k2] == l) { active = false; break; }
        }
        red[tid] = active ? sc[tid] : 0.0f;
    }
    __syncthreads();
    for (int off = KMAX / 2; off > 0; off >>= 1) {
        if (tid < off) red[tid] += red[tid + off];
        __syncthreads();
    }
    const float rowSum = fmaxf(1.0f + red[0], 1e-8f);  // diag contributes 1.0
    __syncthreads();

    if (tid < KMAX) {
        float contrib = 0.0f;
        if (active && sc[tid] > 0.0f) {
            float t = sc[tid] / rowSum;
            float logp = rowp[sl[tid]] * invT - c;     // hits L1/L2 (just streamed)
            contrib = t * (__logf(t) - logp);
        }
        if (tid == 0) {
            float td = 1.0f / rowSum;
            float logp_d = rowp[row] * invT - c;
            contrib += td * (__logf(td) - logp_d);
        }
        red[tid] = contrib;
    }
    __syncthreads();
    for (int off = KMAX / 2; off > 0; off >>= 1) {
        if (tid < off) red[tid] += red[tid + off];
        __syncthreads();
    }
    if (tid == 0) rowLoss[row] = red[0];
}

// ---- final reduction: WMMA ones-vector matmul, one wave32 ------------------
__global__ void __launch_bounds__(32)
final_reduce_kernel(const float* __restrict__ rowLoss, float* __restrict__ out,
                    int Bn) {
    const int lane = threadIdx.x;
    v8f acc = {};
    v2f ones; ones.x = 1.0f; ones.y = 1.0f;

    // check-free full tiles of 64 (B=8192 -> 128 iterations, no tail)
    const int nFull = Bn / 64;
    for (int it = 0; it < nFull; ++it) {
        const float* p = rowLoss + it * 64;
        v2f a;
        a.x = p[lane];
        a.y = p[32 + lane];
        // D[m][n] = sum_k A[m][k] + C[m][n]  (B = ones 4x16)
        acc = __builtin_amdgcn_wmma_f32_16x16x4_f32(
            /*neg_a=*/false, a, /*neg_b=*/false, ones,
            /*c_mod=*/(short)0, acc, /*reuse_a=*/false, /*reuse_b=*/false);
    }
    // scalar tail (empty when Bn % 64 == 0), reduced over the full wave
    float tail = 0.0f;
    for (int i = nFull * 64 + lane; i < Bn; i += 32) tail += rowLoss[i];
    for (int o = 16; o > 0; o >>= 1) tail += __shfl_xor(tail, o, 32);

    // D[m][n] is identical for all n; lanes 0-15 hold M=0..7 in vgprs 0..7,
    // lanes 16-31 hold M=8..15 -> fold the two halves with one shfl_xor.
    float s8 = acc[0] + acc[1] + acc[2] + acc[3] + acc[4] + acc[5] + acc[6] + acc[7];
    float tot = s8 + __shfl_xor(s8, 16, 32) + tail;
    if (lane == 0) out[0] = tot * (TEMP * TEMP / (float)Bn);
}

extern "C" void kernel_launch(void* const* d_in, const int* in_sizes, int n_in,
                              void* d_out, int out_size, void* d_ws, size_t ws_size,
                              hipStream_t stream) {
    const float* logits = (const float*)d_in[0];
    const int*   bidx   = (const int*)d_in[1];
    const int*   tIdx   = (const int*)d_in[2];
    const float* tScore = (const float*)d_in[3];
    const int Bn = in_sizes[1];
    const int K  = in_sizes[2] / Bn;

    int*   g2l     = (int*)d_ws;
    float* rowLoss = (float*)((char*)d_ws + (size_t)NGLOB * sizeof(int));

    g2l_init_kernel<<<(NGLOB + 255) / 256, 256, 0, stream>>>(g2l);
    g2l_scatter_kernel<<<(Bn + 255) / 256, 256, 0, stream>>>(bidx, g2l, Bn);
    row_loss_kernel<<<Bn, 256, 0, stream>>>(logits, tIdx, tScore, g2l, rowLoss, Bn, K);
    final_reduce_kernel<<<1, 32, 0, stream>>>(rowLoss, (float*)d_out, Bn);
}